// MultiheadSelfAttentionwithRelativePositionalEmbedding_26293789786927
// MI455X (gfx1250) — compile-verified
//
#include <hip/hip_runtime.h>
#include <hip/hip_bf16.h>

// ---------------------------------------------------------------------------
// Multihead self-attention with relative positional embeddings (music-
// transformer skew), fp32, on gfx1250 via V_WMMA_F32_16X16X4_F32.
//   B=4, S=1024, DMODEL=512, H=8, HD=64, Lq=2047, Lb=511, Lr=127
// Pipeline:
//   1) qkv_proj_kernel  x3 : X @ W^T + b  -> ws (B,H,S,HD) layout
//   2) scores_kernel        : (q k^T + skew terms)/8 -> attn (in d_out)
//   3) softmax_kernel       : row softmax in place
//   4) av_kernel            : attn @ v -> out (in d_out)
// ---------------------------------------------------------------------------

typedef float v2f __attribute__((ext_vector_type(2)));
typedef float v8f __attribute__((ext_vector_type(8)));

#define BATCH  4
#define SEQ    1024
#define DM     512
#define NHEAD  8
#define HDIM   64
#define LQ     2047
#define LB     511
#define LR     127

static __device__ __forceinline__ v8f wmma4(v2f a, v2f b, v8f c) {
  // D = A(16x4 f32) x B(4x16 f32) + C(16x16 f32)
  return __builtin_amdgcn_wmma_f32_16x16x4_f32(false, a, false, b, (short)0, c,
                                               false, false);
}
static __device__ __forceinline__ int clampi(int x, int lo, int hi) {
  return x < lo ? lo : (x > hi ? hi : x);
}

// ---------------------------------------------------------------------------
// Kernel 1: C[m,n] = X[m,:] . W[n,:] + bias[n], stored as dst[b,h,s,hd].
// grid (8, 256), block 128 (4 waves; each wave owns one 16-wide N tile).
// ---------------------------------------------------------------------------
__global__ __launch_bounds__(128) void qkv_proj_kernel(
    const float* __restrict__ X, const float* __restrict__ W,
    const float* __restrict__ bias, float* __restrict__ dst) {
  const int lane = threadIdx.x & 31;
  const int wave = threadIdx.x >> 5;
  const int half = lane >> 4;   // selects K pair within fragment
  const int q15  = lane & 15;   // M row (A) / N col (B)
  const int n0 = (blockIdx.x * 4 + wave) * 16;
  const int m0 = blockIdx.y * 16;

  const float* arow = X + (size_t)(m0 + q15) * DM + 2 * half;
  const float* brow = W + (size_t)(n0 + q15) * DM + 2 * half;  // B[k,n]=W[n,k]

  v8f acc = {};
#pragma unroll 8
  for (int kk = 0; kk < DM; kk += 4) {
    v2f a = *(const v2f*)(arow + kk);
    v2f b = *(const v2f*)(brow + kk);
    acc = wmma4(a, b, acc);
  }

  const float bn = bias[n0 + q15];
  const int col = n0 + q15;
  const int h = col >> 6, hd = col & 63;
#pragma unroll
  for (int r = 0; r < 8; ++r) {
    const int m = m0 + r + 8 * half;        // D row
    const int b_ = m >> 10, s = m & 1023;   // m = b*S + s
    dst[(((size_t)b_ * NHEAD + h) * SEQ + s) * HDIM + hd] = acc[r] + bn;
  }
}

// ---------------------------------------------------------------------------
// Kernel 2: scores tile = q k^T + quaver/beat/bar relative terms, scaled.
// grid (16, 64, 32), block 128 (wave w handles j-tile blockIdx.x*4+w).
// ---------------------------------------------------------------------------
__global__ __launch_bounds__(128) void scores_kernel(
    const float* __restrict__ Qb, const float* __restrict__ Kb,
    const float* __restrict__ Eq, const float* __restrict__ Eb,
    const float* __restrict__ Er, float* __restrict__ attn) {
  __shared__ float lq[4][16 * 32];    // quaver 16x31 diag staging per wave
  __shared__ float lbr[4][16 * 16];   // beat (cols 0..6) + bar (col 7)

  const int lane = threadIdx.x & 31;
  const int wave = threadIdx.x >> 5;
  const int half = lane >> 4;
  const int q15  = lane & 15;
  const int bh = blockIdx.z;
  const int h  = bh & 7;
  const int i0 = blockIdx.y * 16;
  const int j0 = (blockIdx.x * 4 + wave) * 16;
  const int d  = j0 - i0;             // multiple of 16 (exact /4, /16)

  const float* q  = Qb + (size_t)bh * SEQ * HDIM;
  const float* k  = Kb + (size_t)bh * SEQ * HDIM;
  const float* eq = Eq + (size_t)h * LQ * HDIM;
  const float* eb = Eb + (size_t)h * LB * HDIM;
  const float* er = Er + (size_t)h * LR * HDIM;

  const int l0 = 1008 + d;        // quaver row base: l = l0 + (jl - il + 15)
  const int cb = 255 + d / 4;     // beat center row
  const int cr = 63  + d / 16;    // bar row (constant per tile), in [0,126]

  const float* aptr  = q + (size_t)(i0 + q15) * HDIM + 2 * half;
  const float* kptr  = k + (size_t)(j0 + q15) * HDIM + 2 * half;
  const float* e0ptr = eq + (size_t)clampi(l0 + q15,      0, LQ - 1) * HDIM + 2 * half;
  const float* e1ptr = eq + (size_t)clampi(l0 + 16 + q15, 0, LQ - 1) * HDIM + 2 * half;
  const float* ebptr = (q15 == 7)
      ? (er + (size_t)cr * HDIM + 2 * half)
      : (eb + (size_t)clampi(cb - 3 + q15, 0, LB - 1) * HDIM + 2 * half);

  v8f ak = {}, aq0 = {}, aq1 = {}, ab = {};
#pragma unroll 4
  for (int kk = 0; kk < HDIM; kk += 4) {
    v2f a = *(const v2f*)(aptr + kk);     // shared A fragment (q tile)
    ak  = wmma4(a, *(const v2f*)(kptr + kk),  ak);
    aq0 = wmma4(a, *(const v2f*)(e0ptr + kk), aq0);
    aq1 = wmma4(a, *(const v2f*)(e1ptr + kk), aq1);
    ab  = wmma4(a, *(const v2f*)(ebptr + kk), ab);
  }

  // Stage rel-position tiles in LDS for the diagonal (cross-lane) remap.
#pragma unroll
  for (int r = 0; r < 8; ++r) {
    const int row = r + 8 * half;
    lq[wave][row * 32 + q15]      = aq0[r];
    lq[wave][row * 32 + 16 + q15] = aq1[r];
    lbr[wave][row * 16 + q15]     = ab[r];
  }
  __syncthreads();

  float* arow = attn + ((size_t)bh * SEQ + i0) * SEQ + j0;
#pragma unroll
  for (int r = 0; r < 8; ++r) {
    const int il = r + 8 * half;
    const int jl = q15;
    const int t = jl - il + 15;                  // quaver diag, 0..30
    const int u = (jl + 12 - 4 * (il >> 2)) >> 2; // beat col, 0..6
    float s = ak[r] + lq[wave][il * 32 + t] + lbr[wave][il * 16 + u] +
              lbr[wave][il * 16 + 7];
    arow[(size_t)il * SEQ + jl] = s * 0.125f;    // 1/sqrt(64)
  }
}

// ---------------------------------------------------------------------------
// Kernel 3: row softmax in place over 1024 columns. grid 32768, block 256.
// ---------------------------------------------------------------------------
__global__ __launch_bounds__(256) void softmax_kernel(float* __restrict__ attn) {
  __shared__ float red[256];
  float* p = attn + (size_t)blockIdx.x * SEQ;
  const int tid = threadIdx.x;

  float v0 = p[tid], v1 = p[tid + 256], v2 = p[tid + 512], v3 = p[tid + 768];
  red[tid] = fmaxf(fmaxf(v0, v1), fmaxf(v2, v3));
  __syncthreads();
  for (int s = 128; s > 0; s >>= 1) {
    if (tid < s) red[tid] = fmaxf(red[tid], red[tid + s]);
    __syncthreads();
  }
  const float m = red[0];
  __syncthreads();

  v0 = __expf(v0 - m); v1 = __expf(v1 - m);
  v2 = __expf(v2 - m); v3 = __expf(v3 - m);
  red[tid] = v0 + v1 + v2 + v3;
  __syncthreads();
  for (int s = 128; s > 0; s >>= 1) {
    if (tid < s) red[tid] += red[tid + s];
    __syncthreads();
  }
  const float inv = 1.0f / red[0];
  p[tid] = v0 * inv; p[tid + 256] = v1 * inv;
  p[tid + 512] = v2 * inv; p[tid + 768] = v3 * inv;
}

// ---------------------------------------------------------------------------
// Kernel 4: out = attn @ v, written back to (B,S,DMODEL) layout.
// grid (1, 64, 32), block 128 (4 waves cover hd 0..63).
// ---------------------------------------------------------------------------
__global__ __launch_bounds__(128) void av_kernel(
    const float* __restrict__ attn, const float* __restrict__ Vb,
    float* __restrict__ out) {
  const int lane = threadIdx.x & 31;
  const int wave = threadIdx.x >> 5;
  const int half = lane >> 4;
  const int q15  = lane & 15;
  const int bh = blockIdx.z;
  const int b = bh >> 3, h = bh & 7;
  const int i0 = blockIdx.y * 16;
  const int n0 = wave * 16;

  const float* arow = attn + ((size_t)bh * SEQ + i0 + q15) * SEQ + 2 * half;
  const float* v    = Vb + (size_t)bh * SEQ * HDIM;

  v8f acc = {};
#pragma unroll 4
  for (int kk = 0; kk < SEQ; kk += 4) {
    v2f a = *(const v2f*)(arow + kk);
    v2f bf;
    bf.x = v[(size_t)(kk + 2 * half)     * HDIM + n0 + q15];
    bf.y = v[(size_t)(kk + 2 * half + 1) * HDIM + n0 + q15];
    acc = wmma4(a, bf, acc);
  }
#pragma unroll
  for (int r = 0; r < 8; ++r) {
    const int row = i0 + r + 8 * half;
    out[((size_t)b * SEQ + row) * DM + h * HDIM + n0 + q15] = acc[r];
  }
}

// ---------------------------------------------------------------------------
extern "C" void kernel_launch(void* const* d_in, const int* in_sizes, int n_in,
                              void* d_out, int out_size, void* d_ws,
                              size_t ws_size, hipStream_t stream) {
  const float* query  = (const float*)d_in[0];
  const float* key    = (const float*)d_in[1];
  const float* value  = (const float*)d_in[2];
  const float* Wq     = (const float*)d_in[3];
  const float* bq     = (const float*)d_in[4];
  const float* Wk     = (const float*)d_in[5];
  const float* bk     = (const float*)d_in[6];
  const float* Wv     = (const float*)d_in[7];
  const float* bv     = (const float*)d_in[8];
  const float* quaver = (const float*)d_in[9];
  const float* beat   = (const float*)d_in[10];
  const float* bar    = (const float*)d_in[11];

  const size_t qkvN = (size_t)BATCH * NHEAD * SEQ * HDIM;  // 2,097,152
  float* Q = (float*)d_ws;
  float* K = Q + qkvN;
  float* V = K + qkvN;

  float* out  = (float*)d_out;
  float* attn = out + (size_t)BATCH * SEQ * DM;  // attn region of d_out

  dim3 gProj(8, 256, 1);
  qkv_proj_kernel<<<gProj, 128, 0, stream>>>(query, Wq, bq, Q);
  qkv_proj_kernel<<<gProj, 128, 0, stream>>>(key,   Wk, bk, K);
  qkv_proj_kernel<<<gProj, 128, 0, stream>>>(value, Wv, bv, V);

  scores_kernel<<<dim3(16, 64, 32), 128, 0, stream>>>(Q, K, quaver, beat, bar,
                                                      attn);
  softmax_kernel<<<dim3(BATCH * NHEAD * SEQ), 256, 0, stream>>>(attn);
  av_kernel<<<dim3(1, 64, 32), 128, 0, stream>>>(attn, V, out);
}